// NormLayer_45397804319035
// MI455X (gfx1250) — compile-verified
//
#include <hip/hip_runtime.h>
#include <math.h>

typedef __attribute__((ext_vector_type(2))) float v2f;
typedef __attribute__((ext_vector_type(8))) float v8f;

#define N_ROWS 8192
#define D_DIM  512
#define BM 32                    // query rows per workgroup
#define BN 16                    // keys per tile
#define NWAVES 8                 // 256 threads, wave32
#define DSLICE (D_DIM / NWAVES)  // 64 D-columns per wave
#define NCH (DSLICE / 4)         // 16 K-chunks of 4 for f32 WMMA

// ---------------------------------------------------------------------------
// Pass 1: row L2 normalization  nx[r] = x[r] / max(||x[r]||, 1e-12)
// ---------------------------------------------------------------------------
__global__ __launch_bounds__(256) void rownorm_kernel(const float* __restrict__ x,
                                                      float* __restrict__ nx) {
  __shared__ float red[NWAVES];
  __shared__ float sinv;
  const int row = blockIdx.x;
  const int tid = threadIdx.x;
  const v2f v = *(const v2f*)(x + (size_t)row * D_DIM + tid * 2);
  float p = v.x * v.x + v.y * v.y;
#pragma unroll
  for (int off = 16; off > 0; off >>= 1) p += __shfl_down(p, off, 32);
  if ((tid & 31) == 0) red[tid >> 5] = p;
  __syncthreads();
  if (tid == 0) {
    float s = 0.f;
#pragma unroll
    for (int i = 0; i < NWAVES; ++i) s += red[i];
    sinv = 1.0f / fmaxf(sqrtf(s), 1e-12f);
  }
  __syncthreads();
  const float inv = sinv;
  v2f o;
  o.x = v.x * inv;
  o.y = v.y * inv;
  *(v2f*)(nx + (size_t)row * D_DIM + tid * 2) = o;
}

// ---------------------------------------------------------------------------
// Pass 2: fused masked-softmax attention (flash-attention style)
//   out = 2*x - (softmax(mask(normX @ normX^T)) @ x)
// ---------------------------------------------------------------------------
__global__ __launch_bounds__(256) void contranorm_kernel(const float* __restrict__ x,
                                                         const float* __restrict__ nx,
                                                         const int* __restrict__ adj,
                                                         float* __restrict__ out) {
  __shared__ float sS[BM * BN];  // S tile, then P tile (reused)
  __shared__ float sM[BM];       // running row max
  __shared__ float sL[BM];       // running row sum
  __shared__ float sC[BM];       // per-round correction factor

  const int tid   = threadIdx.x;
  const int wave  = tid >> 5;
  const int lane  = tid & 31;
  const int ln16  = lane & 15;
  const int hi    = lane >> 4;  // 0: lanes 0-15 (K 0,1) | 1: lanes 16-31 (K 2,3)
  const int row0  = blockIdx.x * BM;
  const int dbase = wave * DSLICE;

  // Q fragments (A-matrix, 16x4 f32): resident in VGPRs for the entire sweep.
  v2f aQ[2][NCH];
#pragma unroll
  for (int t = 0; t < 2; ++t)
#pragma unroll
    for (int c = 0; c < NCH; ++c) {
      const int r = row0 + t * 16 + ln16;
      const int k = dbase + c * 4 + 2 * hi;
      aQ[t][c] = *(const v2f*)(nx + (size_t)r * D_DIM + k);
    }

  // O accumulators: 2 M-tiles x 4 N-blocks of 16 columns (this wave's D-slice).
  v8f acc[2][4] = {};

  if (tid < BM) {
    sM[tid] = -1e30f;
    sL[tid] = 0.f;
  }
  __syncthreads();

  for (int j = 0; j < N_ROWS / BN; ++j) {
    const int key0 = j * BN;

    // zero S tile (512 floats / 256 threads)
    sS[tid]       = 0.f;
    sS[tid + 256] = 0.f;
    __syncthreads();

    // ---- S = Q . K^T  (each wave: partial over its 64-wide K-slice) ----
    v8f c0 = {};
    v8f c1 = {};
#pragma unroll
    for (int c = 0; c < NCH; ++c) {
      const int k = dbase + c * 4 + 2 * hi;
      // B-matrix 4x16: lane%16 selects key column n, lane half selects K pair
      const v2f b = *(const v2f*)(nx + (size_t)(key0 + ln16) * D_DIM + k);
      c0 = __builtin_amdgcn_wmma_f32_16x16x4_f32(false, aQ[0][c], false, b,
                                                 (short)0, c0, false, false);
      c1 = __builtin_amdgcn_wmma_f32_16x16x4_f32(false, aQ[1][c], false, b,
                                                 (short)0, c1, false, false);
    }
    // cross-wave reduction of partials into LDS (ds_add_f32)
#pragma unroll
    for (int i = 0; i < 8; ++i) {
      atomicAdd(&sS[(i + hi * 8) * BN + ln16], c0[i]);
      atomicAdd(&sS[(16 + i + hi * 8) * BN + ln16], c1[i]);
    }
    __syncthreads();

    // ---- adjacency mask ----
    // adj is 256 MB streamed exactly once: non-temporal so it never evicts
    // the L2-resident nx/x working set (32 MB << 192 MB L2).
    {
      const int idx = tid * 2;
      const int r   = idx >> 4;
      const int cc  = idx & 15;
      const int* arow = adj + (size_t)(row0 + r) * N_ROWS + key0 + cc;
      const int a0 = __builtin_nontemporal_load(arow);
      const int a1 = __builtin_nontemporal_load(arow + 1);
      if (a0 > 0) sS[idx]     = -__builtin_inff();
      if (a1 > 0) sS[idx + 1] = -__builtin_inff();
    }
    __syncthreads();

    // ---- online softmax row update (one thread per row) ----
    if (tid < BM) {
      float* srow = sS + tid * BN;
      const float mold = sM[tid];
      float mx = mold;
#pragma unroll
      for (int q = 0; q < BN; ++q) mx = fmaxf(mx, srow[q]);
      const float corr = __expf(mold - mx);
      float lsum = 0.f;
#pragma unroll
      for (int q = 0; q < BN; ++q) {
        const float e = __expf(srow[q] - mx);
        srow[q] = e;  // S tile becomes P tile
        lsum += e;
      }
      sM[tid] = mx;
      sL[tid] = sL[tid] * corr + lsum;
      sC[tid] = corr;
    }
    __syncthreads();

    // ---- rescale O, then O += P . V over this wave's D-slice ----
    v2f aP[2][4];
#pragma unroll
    for (int c = 0; c < 4; ++c) {
      aP[0][c] = *(const v2f*)(sS + ln16 * BN + c * 4 + 2 * hi);
      aP[1][c] = *(const v2f*)(sS + (16 + ln16) * BN + c * 4 + 2 * hi);
    }
#pragma unroll
    for (int t = 0; t < 2; ++t) {
#pragma unroll
      for (int i = 0; i < 8; ++i) {
        const float cv = sC[t * 16 + i + hi * 8];
#pragma unroll
        for (int nb = 0; nb < 4; ++nb) acc[t][nb][i] *= cv;
      }
    }
#pragma unroll
    for (int nb = 0; nb < 4; ++nb) {
      const int d = dbase + nb * 16 + ln16;
#pragma unroll
      for (int c = 0; c < 4; ++c) {
        const int key = key0 + c * 4 + 2 * hi;
        v2f b;
        b.x = x[(size_t)key * D_DIM + d];
        b.y = x[(size_t)(key + 1) * D_DIM + d];
        acc[0][nb] = __builtin_amdgcn_wmma_f32_16x16x4_f32(false, aP[0][c], false, b,
                                                           (short)0, acc[0][nb], false, false);
        acc[1][nb] = __builtin_amdgcn_wmma_f32_16x16x4_f32(false, aP[1][c], false, b,
                                                           (short)0, acc[1][nb], false, false);
      }
    }
    __syncthreads();  // sS is re-zeroed at top of next iteration
  }

  // ---- epilogue: out = 2*x - O / l  (output is write-once: non-temporal) ----
#pragma unroll
  for (int t = 0; t < 2; ++t) {
#pragma unroll
    for (int i = 0; i < 8; ++i) {
      const int rl = t * 16 + i + hi * 8;
      const float linv = 1.0f / sL[rl];
      const size_t rowoff = (size_t)(row0 + rl) * D_DIM;
#pragma unroll
      for (int nb = 0; nb < 4; ++nb) {
        const int d = dbase + nb * 16 + ln16;
        const float xv = x[rowoff + d];
        __builtin_nontemporal_store(2.0f * xv - acc[t][nb][i] * linv,
                                    out + rowoff + d);
      }
    }
  }
}

// ---------------------------------------------------------------------------
extern "C" void kernel_launch(void* const* d_in, const int* in_sizes, int n_in,
                              void* d_out, int out_size, void* d_ws, size_t ws_size,
                              hipStream_t stream) {
  (void)in_sizes; (void)n_in; (void)out_size; (void)ws_size;
  const float* x    = (const float*)d_in[0];
  const int*   adj  = (const int*)d_in[1];
  float*       outp = (float*)d_out;
  float*       nx   = (float*)d_ws;  // N*D fp32 = 16 MB workspace

  rownorm_kernel<<<N_ROWS, 256, 0, stream>>>(x, nx);
  contranorm_kernel<<<N_ROWS / BM, 256, 0, stream>>>(x, nx, adj, outp);
}